// SpatialTransformer_50397146251909
// MI455X (gfx1250) — compile-verified
//
#include <hip/hip_runtime.h>
#include <stdint.h>

#define AS1 __attribute__((address_space(1)))
#define AS3 __attribute__((address_space(3)))

typedef int v4i __attribute__((vector_size(16)));

#ifndef __has_builtin
#define __has_builtin(x) 0
#endif

#if __has_builtin(__builtin_amdgcn_global_load_async_to_lds_b128)
#define ST_HAVE_ASYNC_LDS 1
#else
#define ST_HAVE_ASYNC_LDS 0
#endif

#if __has_builtin(__builtin_amdgcn_s_wait_asynccnt)
#define ST_WAIT_ASYNC() __builtin_amdgcn_s_wait_asynccnt(0)
#else
#define ST_WAIT_ASYNC() asm volatile("s_wait_asynccnt 0" ::: "memory")
#endif

namespace {

constexpr int kD0 = 160;
constexpr int kD1 = 192;
constexpr int kD2 = 224;
constexpr int kVox = kD0 * kD1 * kD2;   // 6,881,280 (divisible by 256)
constexpr int kBlock = 256;             // 8 waves (wave32)

__device__ __forceinline__ void axis_interp(float t, int i, int maxv,
                                            int& lo, int& hi,
                                            float& w0, float& w1) {
  // Matches reference: loc clipped to [0,max]; loc0=floor; loc1=min(loc0+1,max);
  // w_c0 = loc1 - loc; w_c1 = 1 - w_c0.
  float l = (float)i + t;
  l = fminf(fmaxf(l, 0.0f), (float)maxv);
  const float lf = floorf(l);
  const float l1 = fminf(lf + 1.0f, (float)maxv);
  lo = (int)lf;
  hi = (int)l1;
  w0 = l1 - l;
  w1 = 1.0f - w0;
}

}  // namespace

__global__ __launch_bounds__(kBlock) void st_trilinear_kernel(
    const float* __restrict__ vol,   // batch 0: [160,192,224,2] f32
    const float* __restrict__ trf,   // batch 0: [160,192,224,3] f32
    float* __restrict__ out) {       // [160,192,224,2] f32
  __shared__ float s_trf[kBlock * 3];  // 3 KB: this block's shift vectors

  const int tid = threadIdx.x;
  const int base = blockIdx.x * kBlock;
  const float* trf_blk = trf + (size_t)base * 3;

  // ---- Stage 3 KB of shift data into LDS: 192 lanes x b128 async copies ----
  if (tid < (kBlock * 3) / 4) {  // 192 lanes * 16 B = 3072 B
#if ST_HAVE_ASYNC_LDS
    __builtin_amdgcn_global_load_async_to_lds_b128(
        (AS1 v4i*)(trf_blk + tid * 4), (AS3 v4i*)(s_trf + tid * 4),
        /*offset=*/0, /*cpol=*/0);
    ST_WAIT_ASYNC();
#else
    const float4 v = ((const float4*)trf_blk)[tid];
    ((float4*)s_trf)[tid] = v;
#endif
  }

  // Warm L2 for the next block's shift stream (24 lanes x 128B line = 3 KB).
  if (tid < 24 && base + kBlock < kVox) {
    __builtin_prefetch(trf_blk + (size_t)kBlock * 3 + tid * 32, 0, 1);
  }

  __syncthreads();

  const int idx = base + tid;
  // Stride-3 word reads: 3*t mod 64 is conflict-free within a wave32.
  const float t0 = s_trf[tid * 3 + 0];
  const float t1 = s_trf[tid * 3 + 1];
  const float t2 = s_trf[tid * 3 + 2];

  const int i2 = idx % kD2;
  const int r  = idx / kD2;
  const int i1 = r % kD1;
  const int i0 = r / kD1;

  int a0, a1, b0, b1, c0, c1;
  float wa0, wa1, wb0, wb1, wc0, wc1;
  axis_interp(t0, i0, kD0 - 1, a0, a1, wa0, wa1);
  axis_interp(t1, i1, kD1 - 1, b0, b1, wb0, wb1);
  axis_interp(t2, i2, kD2 - 1, c0, c1, wc0, wc1);

  // Each corner is a contiguous float2 (C=2) -> global_load_b64 gathers.
  const float2* __restrict__ V = (const float2*)vol;
  const int ra0 = a0 * kD1, ra1 = a1 * kD1;
  const int r00 = (ra0 + b0) * kD2;
  const int r01 = (ra0 + b1) * kD2;
  const int r10 = (ra1 + b0) * kD2;
  const int r11 = (ra1 + b1) * kD2;

  const float2 v000 = V[r00 + c0];
  const float2 v001 = V[r00 + c1];
  const float2 v010 = V[r01 + c0];
  const float2 v011 = V[r01 + c1];
  const float2 v100 = V[r10 + c0];
  const float2 v101 = V[r10 + c1];
  const float2 v110 = V[r11 + c0];
  const float2 v111 = V[r11 + c1];

  const float w000 = wa0 * wb0 * wc0;
  const float w001 = wa0 * wb0 * wc1;
  const float w010 = wa0 * wb1 * wc0;
  const float w011 = wa0 * wb1 * wc1;
  const float w100 = wa1 * wb0 * wc0;
  const float w101 = wa1 * wb0 * wc1;
  const float w110 = wa1 * wb1 * wc0;
  const float w111 = wa1 * wb1 * wc1;

  // Accumulate in the reference's corner order.
  float ox = w000 * v000.x;
  float oy = w000 * v000.y;
  ox = fmaf(w001, v001.x, ox);  oy = fmaf(w001, v001.y, oy);
  ox = fmaf(w010, v010.x, ox);  oy = fmaf(w010, v010.y, oy);
  ox = fmaf(w011, v011.x, ox);  oy = fmaf(w011, v011.y, oy);
  ox = fmaf(w100, v100.x, ox);  oy = fmaf(w100, v100.y, oy);
  ox = fmaf(w101, v101.x, ox);  oy = fmaf(w101, v101.y, oy);
  ox = fmaf(w110, v110.x, ox);  oy = fmaf(w110, v110.y, oy);
  ox = fmaf(w111, v111.x, ox);  oy = fmaf(w111, v111.y, oy);

  ((float2*)out)[idx] = make_float2(ox, oy);  // global_store_b64
}

extern "C" void kernel_launch(void* const* d_in, const int* in_sizes, int n_in,
                              void* d_out, int out_size, void* d_ws, size_t ws_size,
                              hipStream_t stream) {
  (void)in_sizes; (void)n_in; (void)out_size; (void)d_ws; (void)ws_size;
  const float* vol = (const float*)d_in[0];  // batch 0 at offset 0
  const float* trf = (const float*)d_in[1];  // batch 0 at offset 0
  float* out = (float*)d_out;

  const dim3 grid(kVox / kBlock);  // 26,880 blocks, exact
  st_trilinear_kernel<<<grid, dim3(kBlock), 0, stream>>>(vol, trf, out);
}